// Anticipator_81724637708485
// MI455X (gfx1250) — compile-verified
//
#include <hip/hip_runtime.h>
#include <hip/hip_bf16.h>
#include <cstddef>

// ---------------------------------------------------------------------------
// Anticipator forward for MI455X (gfx1250, wave32, WMMA).
// Fixed launch DAG; all data-dependent decisions (priority-queue pops, "best"
// tracking) are resolved on-device into d_ws so the sequence is graph-safe.
// Heavy math: fp32 matvecs with b128 loads (bandwidth bound) and
// V_WMMA_F32_16X16X4_F32 with b64 operand loads for the batched candidate
// projection and the output heads (branch-free inner loop).
// ---------------------------------------------------------------------------

typedef __attribute__((ext_vector_type(2))) float v2f;
typedef __attribute__((ext_vector_type(8))) float v8f;

static constexpr int T_ = 8192, F_ = 2048, H_ = 1024, H4_ = 4096, K3_ = 3072;
static constexpr int NACT = 2513, NVERB = 125, NNOUN = 352;

// ---- workspace layout (float offsets) ----
static constexpr int o_partial = 0;                    // 16*F
static constexpr int o_feat0   = o_partial + 16 * F_;  // F
static constexpr int o_femb    = o_feat0 + F_;         // H
static constexpr int o_h       = o_femb + H_;          // H
static constexpr int o_z       = o_h + H_;             // 4H
static constexpr int o_c       = o_z + H4_;            // H
static constexpr int o_goal    = o_c + H_;             // H
static constexpr int o_xz      = o_goal + H_;          // 4H
static constexpr int o_inp2    = o_xz + H4_;           // 2H
static constexpr int o_a1      = o_inp2 + 2 * H_;      // 10H
static constexpr int o_nf1     = o_a1 + 10 * H_;       // 10H
static constexpr int o_a2      = o_nf1 + 10 * H_;      // 10H
static constexpr int o_nf2     = o_a2 + 10 * H_;       // 10H
static constexpr int o_X       = o_nf2 + 10 * H_;      // 16*3H
static constexpr int o_X2      = o_X + 16 * K3_;       // 16*H
static constexpr int o_dn      = o_X2 + 16 * H_;       // 16
static constexpr int o_dn2     = o_dn + 16;            // 16
static constexpr int o_s1      = o_dn2 + 16;           // 16
static constexpr int o_s2      = o_s1 + 16;            // 16
static constexpr int o_db      = o_s2 + 16;            // 4
static constexpr int o_int     = o_db + 4;             // 16 (int region)

// ---- output layout (float offsets into d_out) ----
static constexpr int oo_act = 0;                       // 2*2513
static constexpr int oo_verb = oo_act + 2 * NACT;      // 2*125
static constexpr int oo_noun = oo_verb + 2 * NVERB;    // 2*352
static constexpr int oo_pf  = oo_noun + 2 * NNOUN;     // 2*1024
static constexpr int oo_pg  = oo_pf + 2 * H_;          // 2*1024
static constexpr int oo_lf  = oo_pg + 2 * H_;          // 1024

// ---------------------------------------------------------------------------
// zero the padded rows of the two WMMA A-operand staging buffers (done each
// call: harness does not re-poison, but we must not rely on leftover state)
__global__ void init_pads(float* __restrict__ X, float* __restrict__ X2) {
  int i = blockIdx.x * blockDim.x + threadIdx.x;
  if (i < 6 * K3_)  X[10 * K3_ + i] = 0.0f;    // rows 10..15 of [16,3072]
  if (i < 14 * H_)  X2[2 * H_ + i] = 0.0f;     // rows 2..15  of [16,1024]
}

// column-max of tsn [T,F]: pass 1 (partial over row chunks, coalesced)
__global__ void colmax_partial(const float* __restrict__ tsn,
                               float* __restrict__ partial, int rowsPer) {
  int c = blockIdx.x * blockDim.x + threadIdx.x;
  int r0 = blockIdx.y * rowsPer;
  float m = -3.402823466e38f;
  for (int r = r0; r < r0 + rowsPer; ++r)
    m = fmaxf(m, tsn[(size_t)r * F_ + c]);
  partial[(size_t)blockIdx.y * F_ + c] = m;
}

__global__ void colmax_final(const float* __restrict__ partial,
                             float* __restrict__ feat0, int nchunks) {
  int c = blockIdx.x * blockDim.x + threadIdx.x;
  float m = -3.402823466e38f;
  for (int i = 0; i < nchunks; ++i) m = fmaxf(m, partial[(size_t)i * F_ + c]);
  feat0[c] = m;
}

// y[row] = dot(W[row, 0:K] (ldw stride), x) + bias[row]? + addend[row]?
// one wave (32 lanes) per output row; b128 weight/vector loads
__global__ void __launch_bounds__(256)
matvec(const float* __restrict__ W, int ldw, const float* __restrict__ x,
       const float* __restrict__ bias, const float* __restrict__ addend,
       float* __restrict__ y, int M, int K) {
  int row = blockIdx.x * 8 + (threadIdx.x >> 5);
  int lane = threadIdx.x & 31;
  if (row >= M) return;
  const float4* wr = (const float4*)(W + (size_t)row * ldw);
  const float4* xv = (const float4*)x;
  float s = 0.0f;
  const int K4 = K >> 2;
  for (int k = lane; k < K4; k += 32) {
    float4 w4 = wr[k];
    float4 x4 = xv[k];
    s += w4.x * x4.x + w4.y * x4.y + w4.z * x4.z + w4.w * x4.w;
  }
  for (int off = 16; off > 0; off >>= 1) s += __shfl_xor(s, off, 32);
  if (lane == 0) {
    if (bias)   s += bias[row];
    if (addend) s += addend[row];
    y[row] = s;
  }
}

// gates z = [i | f | g | o]; c = sig(f)*c_old + sig(i)*tanh(g); h = sig(o)*tanh(c)
__global__ void lstm_pointwise(const float* __restrict__ z,
                               const float* __restrict__ c_in,
                               float* __restrict__ h_out,
                               float* __restrict__ c_out) {
  int t = blockIdx.x * blockDim.x + threadIdx.x;
  if (t >= H_) return;
  float zi = z[t], zf = z[H_ + t], zg = z[2 * H_ + t], zo = z[3 * H_ + t];
  float si = 1.0f / (1.0f + expf(-zi));
  float so = 1.0f / (1.0f + expf(-zo));
  float c = si * tanhf(zg);
  if (c_in) c += (1.0f / (1.0f + expf(-zf))) * c_in[t];
  c_out[t] = c;
  h_out[t] = so * tanhf(c);
}

// assemble A-operand rows: X[i] = [feat | A10[i] | goal], i=0..9
__global__ void buildX(const float* __restrict__ feat,
                       const float* __restrict__ A10,
                       const float* __restrict__ goal,
                       float* __restrict__ X) {
  int idx = blockIdx.x * blockDim.x + threadIdx.x;   // over 10*3H
  int row = idx / K3_;
  int col = idx - row * K3_;
  if (row >= 10) return;
  float v;
  if (col < H_)           v = feat[col];
  else if (col < 2 * H_)  v = A10[(size_t)row * H_ + (col - H_)];
  else                    v = goal[col - 2 * H_];
  X[(size_t)row * K3_ + col] = v;
}

// ---------------------------------------------------------------------------
// WMMA GEMM: Y[r,n] = opt_relu( sum_k X[r,k]*W[n,k] + b[n] ), r<rows_valid,
// n<N. X is [16,K] row-major (padded rows zeroed). One wave per 16-col tile.
// V_WMMA_F32_16X16X4_F32; A/C lane layouts per CDNA5 ISA 7.12.2.
// Branch-free inner loop: lanes with n>=N are clamped to row N-1 and their
// (garbage) columns are masked at store time, so EXEC stays all-1s with no
// per-load predication. Each lane needs K elements {k+2*half, k+2*half+1}
// per step -> single b64 load for A and for B.
__global__ void __launch_bounds__(32)
wmma_xwT(const float* __restrict__ X, const float* __restrict__ W,
         const float* __restrict__ bias, float* __restrict__ Y,
         int K, int N, int rows_valid, int ldY, int relu) {
  const int n0   = blockIdx.x * 16;
  const int lane = threadIdx.x;
  const int half = lane >> 4;        // 0: lanes 0-15, 1: lanes 16-31
  const int m16  = lane & 15;
  const int n    = n0 + m16;
  const bool nvalid = (n < N);
  const int nc = nvalid ? n : (N - 1);             // clamped, always in range
  const float2* wr = (const float2*)(W + (size_t)nc * K);
  const float2* xr = (const float2*)(X + (size_t)m16 * K);

  v8f c = {0.f, 0.f, 0.f, 0.f, 0.f, 0.f, 0.f, 0.f};
  const int steps = K >> 2;                        // K is a multiple of 4
#pragma unroll 4
  for (int s = 0; s < steps; ++s) {
    // A 16x4 f32: lanes 0-15 hold K=0(V0),K=1(V1); lanes 16-31 K=2(V0),K=3(V1)
    float2 av = xr[2 * s + half];
    // B 4x16 f32 (rows striped across lanes): B[kk][n] = W[n*K + kk]
    float2 bv = wr[2 * s + half];
    v2f a; a.x = av.x; a.y = av.y;
    v2f b; b.x = bv.x; b.y = bv.y;
    c = __builtin_amdgcn_wmma_f32_16x16x4_f32(
        /*neg_a=*/false, a, /*neg_b=*/false, b,
        /*c_mod=*/(short)0, c, /*reuse_a=*/false, /*reuse_b=*/false);
  }
  // D: VGPR v holds row (v + 8*half), col = m16
  for (int v = 0; v < 8; ++v) {
    int row = v + 8 * half;
    if (row < rows_valid && nvalid) {
      float outv = c[v] + (bias ? bias[n] : 0.0f);
      if (relu) outv = fmaxf(outv, 0.0f);
      Y[(size_t)row * ldY + n] = outv;
    }
  }
}

// ---------------------------------------------------------------------------
// dn[i] = ||A[i] - goal||^2 / H for 10 rows; optional extra vector (feat_emb)
// handled by the last block -> dn[gridDim.x-1]
__global__ void __launch_bounds__(256)
sqdist_rows(const float* __restrict__ A, const float* __restrict__ goal,
            const float* __restrict__ extra, float* __restrict__ out) {
  __shared__ float red[256];
  int i = blockIdx.x;
  const float* v = (extra && i == (int)gridDim.x - 1) ? extra
                                                      : (A + (size_t)i * H_);
  float s = 0.0f;
  for (int t = threadIdx.x; t < H_; t += 256) {
    float d = v[t] - goal[t];
    s += d * d;
  }
  red[threadIdx.x] = s;
  __syncthreads();
  for (int st = 128; st > 0; st >>= 1) {
    if (threadIdx.x < st) red[threadIdx.x] += red[threadIdx.x + st];
    __syncthreads();
  }
  if (threadIdx.x == 0) out[i] = red[0] / (float)H_;
}

// sequential "best" tracking + first pop: scores of candidates 1..10,
// nstar = first argmin (queue order)
__global__ void chain1(const float* __restrict__ dn /*[0..9]=cand, [10]=D(feat_emb)*/,
                       float* __restrict__ score1, float* __restrict__ db_out,
                       int* __restrict__ nstar_out) {
  float db = dn[10];
  for (int i = 0; i < 10; ++i) {
    score1[i] = db - dn[i];
    if (dn[i] < db) db = dn[i];
  }
  *db_out = db;
  int best = 0;
  float bs = score1[0];
  for (int i = 1; i < 10; ++i)
    if (score1[i] < bs) { bs = score1[i]; best = i; }
  *nstar_out = best;
}

// continue best tracking over expansion 2, then pop the two end nodes in
// queue order ([cand1 \ nstar] then cand2), first-occurrence argmin semantics
__global__ void chain2(const float* __restrict__ dn2,
                       const float* __restrict__ score1,
                       const int* __restrict__ nstar_p,
                       float* __restrict__ score2,
                       int* __restrict__ e0_out, int* __restrict__ e1_out,
                       float* __restrict__ dbslot) {
  float db = *dbslot;
  for (int i = 0; i < 10; ++i) {
    score2[i] = db - dn2[i];
    if (dn2[i] < db) db = dn2[i];
  }
  int ns = *nstar_p;
  int codes[19];
  float sc[19];
  int n = 0;
  for (int i = 0; i < 10; ++i)
    if (i != ns) { codes[n] = i; sc[n] = score1[i]; ++n; }
  for (int i = 0; i < 10; ++i) { codes[n] = 16 + i; sc[n] = score2[i]; ++n; }
  int p0 = 0;
  for (int i = 1; i < 19; ++i)
    if (sc[i] < sc[p0]) p0 = i;
  *e0_out = codes[p0];
  int p1 = (p0 == 0) ? 1 : 0;
  for (int i = 0; i < 19; ++i) {
    if (i == p0) continue;
    if (sc[i] < sc[p1]) p1 = i;
  }
  *e1_out = codes[p1];
}

// copy [a1[nstar] | nf1[nstar]] -> inp2 (device-resolved index)
__global__ void gather_inp2(const float* __restrict__ a1,
                            const float* __restrict__ nf1,
                            const int* __restrict__ nstar,
                            float* __restrict__ inp2) {
  int t = blockIdx.x * blockDim.x + threadIdx.x;
  if (t >= H_) return;
  int ns = *nstar;
  inp2[t]      = a1[(size_t)ns * H_ + t];
  inp2[H_ + t] = nf1[(size_t)ns * H_ + t];
}

// path averaging along parent chains + direct pf/pg/last_feat outputs;
// pa rows staged into X2 for the WMMA head GEMMs
__global__ void assemble(const float* __restrict__ a1, const float* __restrict__ nf1,
                         const float* __restrict__ a2, const float* __restrict__ nf2,
                         const float* __restrict__ feat_emb,
                         const float* __restrict__ goal0,
                         const int* __restrict__ nstar_p,
                         const int* __restrict__ e0_p, const int* __restrict__ e1_p,
                         float* __restrict__ X2, float* __restrict__ out_pf,
                         float* __restrict__ out_pg, float* __restrict__ out_lf) {
  int t = blockIdx.x * blockDim.x + threadIdx.x;
  if (t >= H_) return;
  int ns = *nstar_p;
  int codes0 = *e0_p, codes1 = *e1_p;
  for (int j = 0; j < 2; ++j) {
    int cde = (j == 0) ? codes0 : codes1;
    float pa, pf;
    if (cde < 16) {  // depth-1: path = [child, root(act=0, feat=feat_emb)]
      pa = a1[(size_t)cde * H_ + t] * 0.5f;
      pf = (nf1[(size_t)cde * H_ + t] + feat_emb[t]) * 0.5f;
    } else {         // depth-2: path = [child, nstar, root]
      int cc = cde - 16;
      pa = (a2[(size_t)cc * H_ + t] + a1[(size_t)ns * H_ + t]) * (1.0f / 3.0f);
      pf = (nf2[(size_t)cc * H_ + t] + nf1[(size_t)ns * H_ + t] + feat_emb[t]) *
           (1.0f / 3.0f);
    }
    X2[(size_t)j * H_ + t] = pa;
    out_pf[(size_t)j * H_ + t] = pf;
    out_pg[(size_t)j * H_ + t] = goal0[t];  // all goals on path == goal0
  }
  out_lf[t] = nf1[(size_t)ns * H_ + t];  // last popped node == nstar
}

// ---------------------------------------------------------------------------
extern "C" void kernel_launch(void* const* d_in, const int* in_sizes, int n_in,
                              void* d_out, int out_size, void* d_ws, size_t ws_size,
                              hipStream_t stream) {
  const float* tsn     = (const float*)d_in[0];
  const float* femb_w  = (const float*)d_in[1];
  const float* femb_b  = (const float*)d_in[2];
  const float* g_wih0  = (const float*)d_in[3];
  // d_in[4] = g_whh0: unused (h=0 in single-step goal LSTM)
  const float* g_b0    = (const float*)d_in[5];
  const float* g_wih_r = (const float*)d_in[6];
  // d_in[7] = g_whh_r: unused (h=0)
  const float* g_b_r   = (const float*)d_in[8];
  const float* r_wih   = (const float*)d_in[9];
  const float* r_whh   = (const float*)d_in[10];
  const float* r_b     = (const float*)d_in[11];
  const float* pred_w  = (const float*)d_in[12];
  const float* pred_b  = (const float*)d_in[13];
  const float* act_w   = (const float*)d_in[14];
  const float* act_b   = (const float*)d_in[15];
  const float* verb_w  = (const float*)d_in[16];
  const float* verb_b  = (const float*)d_in[17];
  const float* noun_w  = (const float*)d_in[18];
  const float* noun_b  = (const float*)d_in[19];

  float* ws  = (float*)d_ws;
  float* out = (float*)d_out;
  int*   wsi = (int*)(ws + o_int);

  // 0) init padded GEMM rows
  init_pads<<<72, 256, 0, stream>>>(ws + o_X, ws + o_X2);

  // 1) feat0 = max over T of tsn (coalesced 2-pass)
  colmax_partial<<<dim3(F_ / 256, 16), 256, 0, stream>>>(tsn, ws + o_partial, T_ / 16);
  colmax_final<<<F_ / 256, 256, 0, stream>>>(ws + o_partial, ws + o_feat0, 16);

  // 2) feat_emb = femb_w @ feat0 + femb_b
  matvec<<<H_ / 8, 256, 0, stream>>>(femb_w, F_, ws + o_feat0, femb_b, nullptr,
                                     ws + o_femb, H_, F_);

  // 3) 5-layer goal LSTM, single step, zero states (whh terms vanish)
  //    layer 0: x = [feat_emb | zeros] -> only first H columns of g_wih0
  matvec<<<H4_ / 8, 256, 0, stream>>>(g_wih0, 2 * H_, ws + o_femb, g_b0, nullptr,
                                      ws + o_z, H4_, H_);
  lstm_pointwise<<<H_ / 256, 256, 0, stream>>>(ws + o_z, nullptr, ws + o_h, ws + o_c);
  for (int k = 0; k < 4; ++k) {
    matvec<<<H4_ / 8, 256, 0, stream>>>(g_wih_r + (size_t)k * H4_ * H_, H_,
                                        ws + o_h, g_b_r + (size_t)k * H4_,
                                        nullptr, ws + o_z, H4_, H_);
    float* hout = (k == 3) ? (ws + o_goal) : (ws + o_h);
    lstm_pointwise<<<H_ / 256, 256, 0, stream>>>(ws + o_z, nullptr, hout, ws + o_c);
  }

  // 4) expansion 1: inp = [act0(=0) | feat_emb]; r_wih part constant over steps
  matvec<<<H4_ / 8, 256, 0, stream>>>(r_wih + H_, 2 * H_, ws + o_femb, r_b,
                                      nullptr, ws + o_xz, H4_, H_);
  lstm_pointwise<<<H_ / 256, 256, 0, stream>>>(ws + o_xz, nullptr, ws + o_a1, ws + o_c);
  for (int s = 1; s < 10; ++s) {
    matvec<<<H4_ / 8, 256, 0, stream>>>(r_whh, H_, ws + o_a1 + (size_t)(s - 1) * H_,
                                        nullptr, ws + o_xz, ws + o_z, H4_, H_);
    lstm_pointwise<<<H_ / 256, 256, 0, stream>>>(ws + o_z, ws + o_c,
                                                 ws + o_a1 + (size_t)s * H_, ws + o_c);
  }
  buildX<<<120, 256, 0, stream>>>(ws + o_femb, ws + o_a1, ws + o_goal, ws + o_X);
  wmma_xwT<<<H_ / 16, 32, 0, stream>>>(ws + o_X, pred_w, pred_b, ws + o_nf1,
                                       K3_, H_, 10, H_, 1);
  sqdist_rows<<<11, 256, 0, stream>>>(ws + o_nf1, ws + o_goal, ws + o_femb, ws + o_dn);
  chain1<<<1, 1, 0, stream>>>(ws + o_dn, ws + o_s1, ws + o_db, wsi + 0);

  // 5) expansion 2: node nstar (device-resolved)
  gather_inp2<<<H_ / 256, 256, 0, stream>>>(ws + o_a1, ws + o_nf1, wsi + 0, ws + o_inp2);
  matvec<<<H4_ / 8, 256, 0, stream>>>(r_wih, 2 * H_, ws + o_inp2, r_b, nullptr,
                                      ws + o_xz, H4_, 2 * H_);
  lstm_pointwise<<<H_ / 256, 256, 0, stream>>>(ws + o_xz, nullptr, ws + o_a2, ws + o_c);
  for (int s = 1; s < 10; ++s) {
    matvec<<<H4_ / 8, 256, 0, stream>>>(r_whh, H_, ws + o_a2 + (size_t)(s - 1) * H_,
                                        nullptr, ws + o_xz, ws + o_z, H4_, H_);
    lstm_pointwise<<<H_ / 256, 256, 0, stream>>>(ws + o_z, ws + o_c,
                                                 ws + o_a2 + (size_t)s * H_, ws + o_c);
  }
  buildX<<<120, 256, 0, stream>>>(ws + o_inp2 + H_, ws + o_a2, ws + o_goal, ws + o_X);
  wmma_xwT<<<H_ / 16, 32, 0, stream>>>(ws + o_X, pred_w, pred_b, ws + o_nf2,
                                       K3_, H_, 10, H_, 1);
  sqdist_rows<<<10, 256, 0, stream>>>(ws + o_nf2, ws + o_goal, nullptr, ws + o_dn2);
  chain2<<<1, 1, 0, stream>>>(ws + o_dn2, ws + o_s1, wsi + 0, ws + o_s2,
                              wsi + 1, wsi + 2, ws + o_db);

  // 6) path means + direct pf/pg/last_feat outputs
  assemble<<<H_ / 256, 256, 0, stream>>>(ws + o_a1, ws + o_nf1, ws + o_a2, ws + o_nf2,
                                         ws + o_femb, ws + o_goal, wsi + 0, wsi + 1,
                                         wsi + 2, ws + o_X2, out + oo_pf, out + oo_pg,
                                         out + oo_lf);

  // 7) output heads (M=2 padded to 16) via WMMA
  wmma_xwT<<<(NACT + 15) / 16, 32, 0, stream>>>(ws + o_X2, act_w, act_b,
                                                out + oo_act, H_, NACT, 2, NACT, 0);
  wmma_xwT<<<(NVERB + 15) / 16, 32, 0, stream>>>(ws + o_X2, verb_w, verb_b,
                                                 out + oo_verb, H_, NVERB, 2, NVERB, 0);
  wmma_xwT<<<(NNOUN + 15) / 16, 32, 0, stream>>>(ws + o_X2, noun_w, noun_b,
                                                 out + oo_noun, H_, NNOUN, 2, NNOUN, 0);
}